// ECGSubGraphLayer_5995774345694
// MI455X (gfx1250) — compile-verified
//
#include <hip/hip_runtime.h>
#include <hip/hip_bf16.h>
#include <math.h>

// ---------------------------------------------------------------------------
// MI455X (gfx1250) implementation. fp32 end-to-end using V_WMMA_F32_16X16X4_F32
// for every GEMM-shaped op (QKV, convs-as-GEMM, wave convs, FFN, and a flash
// attention kernel with WMMA for both QK^T and P*V). wave32, 1 wave per tile.
// ---------------------------------------------------------------------------

typedef float v2f __attribute__((ext_vector_type(2)));
typedef float v8f __attribute__((ext_vector_type(8)));

#define WMMA_F32(a, b, c) \
  __builtin_amdgcn_wmma_f32_16x16x4_f32(false, (a), false, (b), (short)0, (c), false, false)

__device__ __forceinline__ float apply_act(float v, int act)
{
  if (act == 1) v = fmaxf(v, 0.0f);
  else if (act == 2) v = 1.0f / (1.0f + __expf(-v));
  return v;
}

// ---------------------------------------------------------------------------
// Fast fp32 WMMA GEMM: C[M,N] = act(A[M,K] * B[K,N] + bias)
// Requirements: N % 64 == 0, M % 32 == 0, K % 4 == 0, B row-major [K,N],
// ldc == N. One wave computes a 32x64 tile: 8 v8f accumulators; B fragments
// are reused across the two 16-row M-subtiles (halves B traffic per FLOP).
// The K-loop is software-pipelined: fragments for step i+1 are prefetched
// before the 8 WMMAs of step i, so the waits become partial (loads overlap
// the matrix pipe instead of s_wait_loadcnt 0 draining before every WMMA).
// ---------------------------------------------------------------------------
__global__ __launch_bounds__(32) void gemm_f32_wmma_fast(
    const float* __restrict__ A, const float* __restrict__ Bm,
    const float* __restrict__ bias, float* __restrict__ C,
    int M, int N, int K, int lda, int act)
{
  const int lane = (int)(threadIdx.x & 31);
  const int half = lane >> 4;
  const int l16  = lane & 15;
  const int m0 = blockIdx.y * 32;
  const int n0 = blockIdx.x * 64;

  // A fragment rows (A layout: lanes 0-15 = rows; lane halves = K pairs).
  const float* arow0 = A + (size_t)(m0 + l16) * lda + 2 * half;
  const float* arow1 = arow0 + (size_t)16 * lda;
  // B fragment pointers (B layout: lane = column, halves = K pairs).
  const float* bp0 = Bm + (size_t)(2 * half) * N + n0 + l16;
  const float* bp1 = bp0 + N;

  v8f acc[8];                       // [mt*4 + nt]
  #pragma unroll
  for (int nt = 0; nt < 4; ++nt) {
    const float bv = (bias != nullptr) ? bias[n0 + nt * 16 + l16] : 0.0f;
    #pragma unroll
    for (int r = 0; r < 8; ++r) { acc[nt][r] = bv; acc[4 + nt][r] = bv; }
  }

  // ---- pipeline prologue: load fragments for k-step 0 ----
  v2f a0 = *(const v2f*)arow0;
  v2f a1 = *(const v2f*)arow1;
  v2f b0, b1, b2, b3;
  b0.x = bp0[0];  b0.y = bp1[0];
  b1.x = bp0[16]; b1.y = bp1[16];
  b2.x = bp0[32]; b2.y = bp1[32];
  b3.x = bp0[48]; b3.y = bp1[48];

  // ---- steady state: prefetch step i+1, then 8 WMMAs on step i ----
  #pragma unroll 2
  for (int kc = 0; kc < K - 4; kc += 4) {
    arow0 += 4;
    arow1 += 4;
    bp0 += (size_t)4 * N;
    bp1 += (size_t)4 * N;
    v2f na0 = *(const v2f*)arow0;
    v2f na1 = *(const v2f*)arow1;
    v2f nb0, nb1, nb2, nb3;
    nb0.x = bp0[0];  nb0.y = bp1[0];
    nb1.x = bp0[16]; nb1.y = bp1[16];
    nb2.x = bp0[32]; nb2.y = bp1[32];
    nb3.x = bp0[48]; nb3.y = bp1[48];

    acc[0] = WMMA_F32(a0, b0, acc[0]);
    acc[1] = WMMA_F32(a0, b1, acc[1]);
    acc[2] = WMMA_F32(a0, b2, acc[2]);
    acc[3] = WMMA_F32(a0, b3, acc[3]);
    acc[4] = WMMA_F32(a1, b0, acc[4]);
    acc[5] = WMMA_F32(a1, b1, acc[5]);
    acc[6] = WMMA_F32(a1, b2, acc[6]);
    acc[7] = WMMA_F32(a1, b3, acc[7]);

    a0 = na0; a1 = na1;
    b0 = nb0; b1 = nb1; b2 = nb2; b3 = nb3;
  }

  // ---- pipeline epilogue: last k-step ----
  acc[0] = WMMA_F32(a0, b0, acc[0]);
  acc[1] = WMMA_F32(a0, b1, acc[1]);
  acc[2] = WMMA_F32(a0, b2, acc[2]);
  acc[3] = WMMA_F32(a0, b3, acc[3]);
  acc[4] = WMMA_F32(a1, b0, acc[4]);
  acc[5] = WMMA_F32(a1, b1, acc[5]);
  acc[6] = WMMA_F32(a1, b2, acc[6]);
  acc[7] = WMMA_F32(a1, b3, acc[7]);

  #pragma unroll
  for (int mt = 0; mt < 2; ++mt)
    #pragma unroll
    for (int nt = 0; nt < 4; ++nt) {
      const int col = n0 + nt * 16 + l16;
      #pragma unroll
      for (int r = 0; r < 8; ++r) {
        const int row = m0 + mt * 16 + r + 8 * half;  // C layout: reg r -> r / r+8
        C[(size_t)row * N + col] = apply_act(acc[mt * 4 + nt][r], act);
      }
    }
}

// ---------------------------------------------------------------------------
// Generic fp32 WMMA GEMM (branchless K-loop), for conv taps / small N:
//   - A row m -> (batch = m / LA, pos = m % LA + shiftA); out-of-range rows
//     are value-masked to zero (address clamped, load unconditional).
//   - Out-of-range B columns only need ADDRESS clamping: WMMA column n of D
//     depends only on column n of B, and those columns are never stored.
//   - B element (k,n) at Bm[k*ldbk + n*ldbn] (conv [Cout][Cin][K] layouts).
// ---------------------------------------------------------------------------
__global__ __launch_bounds__(32) void gemm_f32_wmma_gen(
    const float* __restrict__ A, const float* __restrict__ Bm,
    const float* __restrict__ bias, float* __restrict__ C,
    int M, int N, int K, int lda, int ldbk, int ldbn, int ldc,
    int LA, int shiftA, int accumulate, int act)
{
  const int lane = (int)(threadIdx.x & 31);
  const int half = lane >> 4;
  const int l16  = lane & 15;
  const int m0 = blockIdx.y * 16;
  const int n0 = blockIdx.x * 64;

  const int rowA = m0 + l16;
  const int ab = rowA / LA;
  int al = rowA - ab * LA + shiftA;
  const float amask = (al >= 0 && al < LA) ? 1.0f : 0.0f;
  al = min(max(al, 0), LA - 1);
  const float* arow = A + ((size_t)ab * LA + (size_t)al) * lda + 2 * half;

  // Clamped B column pointers.
  const float* bp[4];
  #pragma unroll
  for (int nt = 0; nt < 4; ++nt) {
    const int colc = min(n0 + nt * 16 + l16, N - 1);
    bp[nt] = Bm + (size_t)(2 * half) * ldbk + (size_t)colc * ldbn;
  }

  v8f acc[4];
  #pragma unroll
  for (int nt = 0; nt < 4; ++nt) {
    const int colc = min(n0 + nt * 16 + l16, N - 1);
    if (accumulate) {
      #pragma unroll
      for (int r = 0; r < 8; ++r)
        acc[nt][r] = C[(size_t)(m0 + r + 8 * half) * ldc + colc];
    } else {
      const float bv = (bias != nullptr) ? bias[colc] : 0.0f;
      #pragma unroll
      for (int r = 0; r < 8; ++r) acc[nt][r] = bv;
    }
  }

  #pragma unroll 2
  for (int kc = 0; kc < K; kc += 4) {
    v2f a = *(const v2f*)arow;
    a.x *= amask;
    a.y *= amask;
    #pragma unroll
    for (int nt = 0; nt < 4; ++nt) {
      v2f b;
      b.x = bp[nt][0];
      b.y = bp[nt][ldbk];
      acc[nt] = WMMA_F32(a, b, acc[nt]);
      bp[nt] += (size_t)4 * ldbk;
    }
    arow += 4;
  }

  #pragma unroll
  for (int nt = 0; nt < 4; ++nt) {
    const int col = n0 + nt * 16 + l16;
    if (col >= N) continue;
    #pragma unroll
    for (int r = 0; r < 8; ++r) {
      const int row = m0 + r + 8 * half;
      C[(size_t)row * ldc + col] = apply_act(acc[nt][r], act);
    }
  }
}

// ---------------------------------------------------------------------------
// Softmax over W=3 wave logits -> wave_mask, argmax -> wave-type embedding add.
// ---------------------------------------------------------------------------
__global__ __launch_bounds__(128) void mask_emb_kernel(
    const float* __restrict__ x, const float* __restrict__ wp,
    const float* __restrict__ emb, float* __restrict__ mask_out,
    float* __restrict__ x1, int D)
{
  const size_t row = blockIdx.x;
  const int t = (int)threadIdx.x;
  const float w0 = wp[row * 3 + 0];
  const float w1 = wp[row * 3 + 1];
  const float w2 = wp[row * 3 + 2];
  const float mx = fmaxf(w0, fmaxf(w1, w2));
  const float e0 = __expf(w0 - mx), e1 = __expf(w1 - mx), e2 = __expf(w2 - mx);
  const float inv = 1.0f / (e0 + e1 + e2);
  const float m0 = e0 * inv, m1 = e1 * inv, m2 = e2 * inv;
  if (t == 0) {
    mask_out[row * 3 + 0] = m0;
    mask_out[row * 3 + 1] = m1;
    mask_out[row * 3 + 2] = m2;
  }
  const int lbl = (m0 >= m1 && m0 >= m2) ? 0 : ((m1 >= m2) ? 1 : 2);
  for (int d = t; d < D; d += 128)
    x1[row * D + d] = x[row * D + d] + emb[(size_t)lbl * D + d];
}

// wave_sim[b,q,k] = <mask[b,q,:], mask[b,k,:]>  (K=3 -> VALU; 50 MFLOP)
__global__ __launch_bounds__(256) void wave_sim_kernel(
    const float* __restrict__ mask, float* __restrict__ sim, int L)
{
  const size_t idx = (size_t)blockIdx.x * blockDim.x + threadIdx.x;
  const int k = (int)(idx % L);
  const size_t tq = idx / L;
  const int q = (int)(tq % L);
  const int b = (int)(tq / L);
  const float* mq = mask + ((size_t)b * L + q) * 3;
  const float* mk = mask + ((size_t)b * L + k) * 3;
  sim[idx] = mq[0] * mk[0] + mq[1] * mk[1] + mq[2] * mk[2];
}

// ---------------------------------------------------------------------------
// fp32 flash attention, one wave per 16-query tile per (b,h).
//   scores = (QK^T)/sqrt(hd);  s' = s*(gate[k] + sim[q,k]) + sim[q,k]
//   online softmax over k, P*V accumulated via WMMA (P transposed via LDS).
// ---------------------------------------------------------------------------
__global__ __launch_bounds__(32) void flash_attn_f32(
    const float* __restrict__ Q, const float* __restrict__ Km,
    const float* __restrict__ V, const float* __restrict__ G,
    const float* __restrict__ Sim, float* __restrict__ O,
    int B, int L, int H, int D)
{
  __shared__ float lds[16 * 16];
  const int lane = (int)(threadIdx.x & 31);
  const int half = lane >> 4;
  const int l16  = lane & 15;
  const int q0 = blockIdx.x * 16;
  const int h  = blockIdx.y;
  const int b  = blockIdx.z;
  const int hd = D / H;                 // 64
  const float scale = rsqrtf((float)hd);

  // Q fragments (A layout), 16 chunks of K=4 over hd=64.
  const float* qrow = Q + ((size_t)(b * L + q0 + l16)) * D + h * hd;
  v2f qa[16];
  #pragma unroll
  for (int kc = 0; kc < 16; ++kc) qa[kc] = *(const v2f*)(qrow + kc * 4 + 2 * half);

  float mrun[8], srun[8];
  v8f o[4];
  #pragma unroll
  for (int r = 0; r < 8; ++r) { mrun[r] = -3.0e38f; srun[r] = 0.0f; }
  #pragma unroll
  for (int nt = 0; nt < 4; ++nt)
    #pragma unroll
    for (int r = 0; r < 8; ++r) o[nt][r] = 0.0f;

  for (int kt = 0; kt < L; kt += 16) {
    // S = Q * K^T : loading K tile rows in "A addressing" yields B fragments
    // of K^T (B layout: lane = N(key), lane-half selects K(dim) pairs).
    const float* krow = Km + ((size_t)(b * L + kt + l16)) * D + h * hd;
    v8f s;
    #pragma unroll
    for (int r = 0; r < 8; ++r) s[r] = 0.0f;
    #pragma unroll
    for (int kc = 0; kc < 16; ++kc) {
      v2f kb = *(const v2f*)(krow + kc * 4 + 2 * half);
      s = WMMA_F32(qa[kc], kb, s);
    }

    const float gk = G[(size_t)(b * L + kt + l16) * H + h];     // gate at key k
    const float* simcol = Sim + ((size_t)b * L + q0) * L + kt + l16;

    #pragma unroll
    for (int r = 0; r < 8; ++r) {
      const int qr = r + 8 * half;                              // local q row
      const float sv = simcol[(size_t)qr * L];
      float xs = s[r] * scale * (gk + sv) + sv;
      // row-max across the 16 lanes holding this row
      float mx = xs;
      #pragma unroll
      for (int off = 1; off < 16; off <<= 1) mx = fmaxf(mx, __shfl_xor(mx, off));
      const float mnew = fmaxf(mrun[r], mx);
      const float corr = __expf(mrun[r] - mnew);
      const float p = __expf(xs - mnew);
      float ps = p;
      #pragma unroll
      for (int off = 1; off < 16; off <<= 1) ps += __shfl_xor(ps, off);
      srun[r] = srun[r] * corr + ps;
      mrun[r] = mnew;
      #pragma unroll
      for (int nt = 0; nt < 4; ++nt) o[nt][r] *= corr;
      s[r] = p;
    }

    // Transpose P (C layout -> A layout) through LDS.
    __syncthreads();
    #pragma unroll
    for (int r = 0; r < 8; ++r) lds[(r + 8 * half) * 16 + l16] = s[r];
    __syncthreads();

    // O += P * V  (P: 16x16, V tile: 16x64)
    #pragma unroll
    for (int kc = 0; kc < 4; ++kc) {
      v2f pa;
      pa.x = lds[l16 * 16 + kc * 4 + 2 * half];
      pa.y = lds[l16 * 16 + kc * 4 + 2 * half + 1];
      const int kk = kt + kc * 4 + 2 * half;
      #pragma unroll
      for (int nt = 0; nt < 4; ++nt) {
        const float* vp = V + ((size_t)(b * L + kk)) * D + h * hd + nt * 16 + l16;
        v2f vb;
        vb.x = vp[0];
        vb.y = vp[(size_t)D];
        o[nt] = WMMA_F32(pa, vb, o[nt]);
      }
    }
  }

  #pragma unroll
  for (int nt = 0; nt < 4; ++nt)
    #pragma unroll
    for (int r = 0; r < 8; ++r) {
      const int row = q0 + r + 8 * half;
      O[((size_t)(b * L + row)) * D + h * hd + nt * 16 + l16] = o[nt][r] / srun[r];
    }
}

// ---------------------------------------------------------------------------
// LayerNorm helpers (D = 512, 256 threads, 2 elements / thread)
// ---------------------------------------------------------------------------
__device__ inline float block_sum256(float v, float* sm)
{
  const int t = (int)threadIdx.x;
  sm[t] = v;
  __syncthreads();
  #pragma unroll
  for (int s = 128; s > 0; s >>= 1) {
    if (t < s) sm[t] += sm[t + s];
    __syncthreads();
  }
  const float r = sm[0];
  __syncthreads();
  return r;
}

// out = LN2( res + LN1(a + res) )
__global__ __launch_bounds__(256) void double_ln_kernel(
    const float* __restrict__ A, const float* __restrict__ R,
    const float* __restrict__ g1, const float* __restrict__ b1,
    const float* __restrict__ g2, const float* __restrict__ b2,
    float* __restrict__ Out, int D)
{
  __shared__ float sm[256];
  const size_t row = blockIdx.x;
  const int t = (int)threadIdx.x;
  float v[2], res[2];
  #pragma unroll
  for (int i = 0; i < 2; ++i) {
    const int d = t + i * 256;
    res[i] = R[row * D + d];
    v[i] = A[row * D + d] + res[i];
  }
  float mean = block_sum256(v[0] + v[1], sm) / (float)D;
  float var = block_sum256((v[0] - mean) * (v[0] - mean) +
                           (v[1] - mean) * (v[1] - mean), sm) / (float)D;
  float inv = rsqrtf(var + 1e-5f);
  float z[2];
  #pragma unroll
  for (int i = 0; i < 2; ++i) {
    const int d = t + i * 256;
    z[i] = res[i] + (v[i] - mean) * inv * g1[d] + b1[d];
  }
  mean = block_sum256(z[0] + z[1], sm) / (float)D;
  var = block_sum256((z[0] - mean) * (z[0] - mean) +
                     (z[1] - mean) * (z[1] - mean), sm) / (float)D;
  inv = rsqrtf(var + 1e-5f);
  #pragma unroll
  for (int i = 0; i < 2; ++i) {
    const int d = t + i * 256;
    Out[row * D + d] = (z[i] - mean) * inv * g2[d] + b2[d];
  }
}

// conv_out = relu(sum_w feats_w * mask_w * ww_w); out = LN2(x + LN1(conv_out))
__global__ __launch_bounds__(256) void combine_ln_kernel(
    const float* __restrict__ F0, const float* __restrict__ F1,
    const float* __restrict__ F2, const float* __restrict__ mask,
    const float* __restrict__ ww, const float* __restrict__ X,
    const float* __restrict__ g1, const float* __restrict__ b1,
    const float* __restrict__ g2, const float* __restrict__ b2,
    float* __restrict__ Out, int D)
{
  __shared__ float sm[256];
  const size_t row = blockIdx.x;
  const int t = (int)threadIdx.x;
  const float m0 = mask[row * 3 + 0] * ww[0];
  const float m1 = mask[row * 3 + 1] * ww[1];
  const float m2 = mask[row * 3 + 2] * ww[2];
  float v[2], res[2];
  #pragma unroll
  for (int i = 0; i < 2; ++i) {
    const int d = t + i * 256;
    const float c = F0[row * D + d] * m0 + F1[row * D + d] * m1 + F2[row * D + d] * m2;
    v[i] = fmaxf(c, 0.0f);
    res[i] = X[row * D + d];
  }
  float mean = block_sum256(v[0] + v[1], sm) / (float)D;
  float var = block_sum256((v[0] - mean) * (v[0] - mean) +
                           (v[1] - mean) * (v[1] - mean), sm) / (float)D;
  float inv = rsqrtf(var + 1e-5f);
  float z[2];
  #pragma unroll
  for (int i = 0; i < 2; ++i) {
    const int d = t + i * 256;
    z[i] = res[i] + (v[i] - mean) * inv * g1[d] + b1[d];
  }
  mean = block_sum256(z[0] + z[1], sm) / (float)D;
  var = block_sum256((z[0] - mean) * (z[0] - mean) +
                     (z[1] - mean) * (z[1] - mean), sm) / (float)D;
  inv = rsqrtf(var + 1e-5f);
  #pragma unroll
  for (int i = 0; i < 2; ++i) {
    const int d = t + i * 256;
    Out[row * D + d] = (z[i] - mean) * inv * g2[d] + b2[d];
  }
}

// out = LN(a + res)
__global__ __launch_bounds__(256) void add_ln_kernel(
    const float* __restrict__ A, const float* __restrict__ R,
    const float* __restrict__ g, const float* __restrict__ b,
    float* __restrict__ Out, int D)
{
  __shared__ float sm[256];
  const size_t row = blockIdx.x;
  const int t = (int)threadIdx.x;
  float v[2];
  #pragma unroll
  for (int i = 0; i < 2; ++i) {
    const int d = t + i * 256;
    v[i] = A[row * D + d] + R[row * D + d];
  }
  float mean = block_sum256(v[0] + v[1], sm) / (float)D;
  float var = block_sum256((v[0] - mean) * (v[0] - mean) +
                           (v[1] - mean) * (v[1] - mean), sm) / (float)D;
  float inv = rsqrtf(var + 1e-5f);
  #pragma unroll
  for (int i = 0; i < 2; ++i) {
    const int d = t + i * 256;
    Out[row * D + d] = (v[i] - mean) * inv * g[d] + b[d];
  }
}

// ---------------------------------------------------------------------------
extern "C" void kernel_launch(void* const* d_in, const int* in_sizes, int n_in,
                              void* d_out, int out_size, void* d_ws, size_t ws_size,
                              hipStream_t stream)
{
  (void)in_sizes; (void)n_in; (void)out_size; (void)ws_size;

  const int Bn = 8, L = 1024, D = 512, H = 8, W = 3, HID = 64, G1 = 128, DF = 2048;
  const int BL = Bn * L;   // 8192

  const float* x        = (const float*)d_in[0];
  const float* conv1_w  = (const float*)d_in[1];
  const float* conv1_b  = (const float*)d_in[2];
  const float* conv2_w  = (const float*)d_in[3];
  const float* conv2_b  = (const float*)d_in[4];
  const float* conv3_w  = (const float*)d_in[5];
  const float* conv3_b  = (const float*)d_in[6];
  const float* wave_emb = (const float*)d_in[7];
  const float* wq_w = (const float*)d_in[8];  const float* wq_b = (const float*)d_in[9];
  const float* wk_w = (const float*)d_in[10]; const float* wk_b = (const float*)d_in[11];
  const float* wv_w = (const float*)d_in[12]; const float* wv_b = (const float*)d_in[13];
  const float* g1_w = (const float*)d_in[14]; const float* g1_b = (const float*)d_in[15];
  const float* g2_w = (const float*)d_in[16]; const float* g2_b = (const float*)d_in[17];
  const float* ln_attn_g = (const float*)d_in[18]; const float* ln_attn_b = (const float*)d_in[19];
  const float* norm1_g   = (const float*)d_in[20]; const float* norm1_b   = (const float*)d_in[21];
  const float* ln_conv_g = (const float*)d_in[22]; const float* ln_conv_b = (const float*)d_in[23];
  const float* norm2_g   = (const float*)d_in[24]; const float* norm2_b   = (const float*)d_in[25];
  const float* norm3_g   = (const float*)d_in[26]; const float* norm3_b   = (const float*)d_in[27];
  const float* wc_w = (const float*)d_in[28]; const float* wc_b = (const float*)d_in[29];
  const float* wave_weights = (const float*)d_in[30];
  const float* f1_w = (const float*)d_in[31]; const float* f1_b = (const float*)d_in[32];
  const float* f2_w = (const float*)d_in[33]; const float* f2_b = (const float*)d_in[34];

  // Output layout: x [B,L,D], wave_mask [B,L,W], subgraph_adj [B,L,L]
  float* out_x    = (float*)d_out;
  float* out_mask = out_x + (size_t)BL * D;
  float* out_adj  = out_mask + (size_t)BL * W;

  // Workspace layout (~177 MB of fp32), with reuse after attention.
  float* ws   = (float*)d_ws;
  float* x1   = ws;                         // [BL, D]
  float* qb   = x1   + (size_t)BL * D;      // [BL, D]   later: feats_w0
  float* kb   = qb   + (size_t)BL * D;      // [BL, D]   later: feats_w1
  float* vb   = kb   + (size_t)BL * D;      // [BL, D]   later: feats_w2
  float* attn = vb   + (size_t)BL * D;      // [BL, D]   later: x3
  float* x2   = attn + (size_t)BL * D;      // [BL, D]
  float* h1   = x2   + (size_t)BL * D;      // [BL, HID]
  float* h2   = h1   + (size_t)BL * HID;    // [BL, HID]
  float* wp   = h2   + (size_t)BL * HID;    // [BL, 3] (pad 4)
  float* gates= wp   + (size_t)BL * 4;      // [BL, H]
  float* g1o  = gates+ (size_t)BL * H;      // [BL, 128]
  float* ffn1 = g1o  + (size_t)BL * G1;     // [BL, 2048]

  const dim3 blk32(32), blk256(256), blk128(128);

  // --- WaveDetector conv1 (D->64, K=5, pad 2) as 5 accumulating GEMM taps ---
  for (int t = 0; t < 5; ++t)
    gemm_f32_wmma_gen<<<dim3(1, BL / 16), blk32, 0, stream>>>(
        x, conv1_w + t, (t == 0) ? conv1_b : nullptr, h1,
        BL, HID, D, D, /*ldbk=*/5, /*ldbn=*/D * 5, HID,
        L, t - 2, (t > 0) ? 1 : 0, (t == 4) ? 1 : 0);

  // --- conv2 (64->64, K=3, pad 1) ---
  for (int t = 0; t < 3; ++t)
    gemm_f32_wmma_gen<<<dim3(1, BL / 16), blk32, 0, stream>>>(
        h1, conv2_w + t, (t == 0) ? conv2_b : nullptr, h2,
        BL, HID, HID, HID, /*ldbk=*/3, /*ldbn=*/HID * 3, HID,
        L, t - 1, (t > 0) ? 1 : 0, (t == 2) ? 1 : 0);

  // --- conv3 (64->3, K=3, pad 1) ---
  for (int t = 0; t < 3; ++t)
    gemm_f32_wmma_gen<<<dim3(1, BL / 16), blk32, 0, stream>>>(
        h2, conv3_w + t, (t == 0) ? conv3_b : nullptr, wp,
        BL, W, HID, HID, /*ldbk=*/3, /*ldbn=*/HID * 3, W,
        L, t - 1, (t > 0) ? 1 : 0, 0);

  // --- softmax over W -> wave_mask; argmax -> embedding add -> x1 ---
  mask_emb_kernel<<<BL, blk128, 0, stream>>>(x, wp, wave_emb, out_mask, x1, D);

  // --- wave_sim == subgraph_adj, written straight to output ---
  wave_sim_kernel<<<(Bn * L * L) / 256, blk256, 0, stream>>>(out_mask, out_adj, L);

  // --- QKV projections (fast path: 32x64 tiles, software-pipelined) ---
  gemm_f32_wmma_fast<<<dim3(D / 64, BL / 32), blk32, 0, stream>>>(
      x1, wq_w, wq_b, qb, BL, D, D, D, 0);
  gemm_f32_wmma_fast<<<dim3(D / 64, BL / 32), blk32, 0, stream>>>(
      x1, wk_w, wk_b, kb, BL, D, D, D, 0);
  gemm_f32_wmma_fast<<<dim3(D / 64, BL / 32), blk32, 0, stream>>>(
      x1, wv_w, wv_b, vb, BL, D, D, D, 0);

  // --- gate MLP: relu(x1@g1) then sigmoid(@g2) ---
  gemm_f32_wmma_fast<<<dim3(G1 / 64, BL / 32), blk32, 0, stream>>>(
      x1, g1_w, g1_b, g1o, BL, G1, D, D, 1);
  gemm_f32_wmma_gen<<<dim3(1, BL / 16), blk32, 0, stream>>>(
      g1o, g2_w, g2_b, gates, BL, H, G1, G1, H, 1, H, L, 0, 0, 2);

  // --- flash attention with gate/wave_sim score modification ---
  flash_attn_f32<<<dim3(L / 16, H, Bn), blk32, 0, stream>>>(
      qb, kb, vb, gates, out_adj, attn, Bn, L, H, D);

  // --- x2 = norm1(x1 + ln_attn(attn + x1)) ---
  double_ln_kernel<<<BL, blk256, 0, stream>>>(
      attn, x1, ln_attn_g, ln_attn_b, norm1_g, norm1_b, x2, D);

  // --- per-wave-type 512x512 convs (Q/K/V buffers recycled as feats) ---
  float* feats[3] = { qb, kb, vb };
  for (int w = 0; w < 3; ++w)
    gemm_f32_wmma_fast<<<dim3(D / 64, BL / 32), blk32, 0, stream>>>(
        x2, wc_w + (size_t)w * D * D, wc_b + (size_t)w * D, feats[w],
        BL, D, D, D, 0);

  // --- x3 = norm2(x2 + ln_conv(relu(sum_w feats*mask*ww)))  (x3 := attn buf) ---
  combine_ln_kernel<<<BL, blk256, 0, stream>>>(
      qb, kb, vb, out_mask, wave_weights, x2,
      ln_conv_g, ln_conv_b, norm2_g, norm2_b, attn, D);

  // --- FFN ---
  gemm_f32_wmma_fast<<<dim3(DF / 64, BL / 32), blk32, 0, stream>>>(
      attn, f1_w, f1_b, ffn1, BL, DF, D, D, 1);
  gemm_f32_wmma_fast<<<dim3(D / 64, BL / 32), blk32, 0, stream>>>(
      ffn1, f2_w, f2_b, x1, BL, D, DF, DF, 0);

  // --- final: out_x = norm3(x3 + ffn_out) ---
  add_ln_kernel<<<BL, blk256, 0, stream>>>(
      x1, attn, norm3_g, norm3_b, out_x, D);
}